// PositionalEncoder_34419867910572
// MI455X (gfx1250) — compile-verified
//
#include <hip/hip_runtime.h>
#include <stdint.h>

typedef float v2f __attribute__((ext_vector_type(2)));
typedef float v4f __attribute__((ext_vector_type(4)));
typedef float v8f __attribute__((ext_vector_type(8)));

#define TILE 512      // output positions per workgroup
#define NCH  8        // channels, interleaved innermost

#if defined(__HIP_DEVICE_COMPILE__) && __has_builtin(__builtin_amdgcn_wmma_f32_16x16x4_f32)
#define USE_WMMA 1
#else
#define USE_WMMA 0
#endif

// sym5 decomposition filters (lo), hi[k] = (-1)^(k+1) * lo[9-k]
__device__ __constant__ float c_LO[10] = {
     0.027333068345077982f,  0.029519490925774643f, -0.039134249302383094f,
     0.1993975339773936f,    0.7234076904024206f,    0.6339789634582119f,
     0.01660210576452232f,  -0.17532808990845047f,  -0.021101834024758855f,
     0.019538882735286728f };
__device__ __constant__ float c_HI[10] = {
    -0.019538882735286728f, -0.021101834024758855f,  0.17532808990845047f,
     0.01660210576452232f,  -0.6339789634582119f,    0.7234076904024206f,
    -0.1993975339773936f,   -0.039134249302383094f, -0.029519490925774643f,
     0.027333068345077982f };

// CDNA5 async tensor-path copy: global -> LDS, tracked by ASYNCcnt.
__device__ __forceinline__ void async_ld_b128(unsigned lds_byte_off, const float* g) {
    asm volatile("global_load_async_to_lds_b128 %0, %1, off"
                 :: "v"(lds_byte_off), "v"(g)
                 : "memory");
}
__device__ __forceinline__ void wait_async0() {
    asm volatile("s_wait_asynccnt 0" ::: "memory");
}

__device__ __forceinline__ int mirror_idx(int t, int n) {
    t = (t < 0) ? (-t - 1) : t;
    t = (t >= n) ? (2 * n - 1 - t) : t;
    return t;
}

// Streaming copy of orig -> front of output (touched once: non-temporal).
__global__ void copy_orig_kernel(const v4f* __restrict__ src,
                                 v4f* __restrict__ dst, long n4) {
    long i = (long)blockIdx.x * blockDim.x + threadIdx.x;
    const long stride = (long)gridDim.x * blockDim.x;
    for (; i < n4; i += stride) {
        v4f v = __builtin_nontemporal_load(&src[i]);
        __builtin_nontemporal_store(v, &dst[i]);
    }
}

// One DWT level. in: [n][8] interleaved. outA/outD: [nout][8] interleaved.
// y_f[m] = sum_tap h_f[tap] * s[mirror(2m+1-tap)]   (pywt 'symmetric')
__global__ void dwt_level_kernel(const float* __restrict__ in, int n,
                                 float* __restrict__ outA,
                                 float* __restrict__ outD, int nout) {
    __shared__ __attribute__((aligned(16))) float smem[(2 * TILE + 24) * NCH + 8];
    __shared__ __attribute__((aligned(16))) float stage[8][2 * 8 * NCH]; // per-wave

    const int tileM0 = blockIdx.x * TILE;
    int mEnd = tileM0 + TILE; if (mEnd > nout) mEnd = nout;

    // Global sample span needed by this tile (all mirrored indices land inside).
    const int tmax = 2 * (mEnd - 1) + 1;
    int lo = 2 * tileM0 - 8;
    if (tmax > n - 1) { int ml = 2 * n - 1 - tmax; if (ml < lo) lo = ml; }
    if (lo < 0) lo = 0;
    int hi = tmax; if (hi > n - 1) hi = n - 1;
    const int span   = hi - lo + 1;
    const int nchunk = span * 2;                       // 16B chunks (span*8 floats)

    // Stage the window into LDS with async tensor-path loads (ASYNCcnt).
    const float* gbase = in + (size_t)lo * NCH;
    const unsigned sbase = (unsigned)(uintptr_t)(&smem[0]);   // low 32b = LDS offset
    for (int c = threadIdx.x; c < nchunk; c += blockDim.x) {
        async_ld_b128(sbase + (unsigned)c * 16u, gbase + (size_t)c * 4);
    }
    wait_async0();
    __syncthreads();

#if USE_WMMA
    // Polyphase WMMA: D(16x8 used) = A(16 rows = 8 positions x 2 filters, K=24)
    //                              x B(K=24 samples, 8 channel columns), 6x K=4.
    // B columns 8-15 are don't-care (column-separable) -> lanes 8-15 mirror ch&7.
    const int lane = threadIdx.x & 31;
    const int wav  = threadIdx.x >> 5;
    const int nwav = blockDim.x >> 5;
    const int col  = lane & 15;        // A row r=2p+f | B/D column (channel)
    const int colm = col & 7;          // effective channel for B gather
    const int khi  = lane >> 4;        // K half: lanes 16-31 hold K=+2,+3 / M=+8

    // Constant A: row r = 2p+f, A[r,rel] = h_f[2p+9-rel] (else 0).
    v2f Amat[6];
    {
        const int p = col >> 1, f = col & 1;
#pragma unroll
        for (int c = 0; c < 6; ++c) {
            const int k0 = 4 * c + 2 * khi;
            const int t0 = 2 * p + 9 - k0;        // tap for K=k0
            const int t1 = t0 - 1;                // tap for K=k0+1
            Amat[c].x = (t0 >= 0 && t0 <= 9) ? (f ? c_HI[t0] : c_LO[t0]) : 0.f;
            Amat[c].y = (t1 >= 0 && t1 <= 9) ? (f ? c_HI[t1] : c_LO[t1]) : 0.f;
        }
    }

    const int ngroups = (mEnd - tileM0 + 7) >> 3;  // 8 output positions per group
    const int spanMax = span - 1;
    for (int g = wav; g < ngroups; g += nwav) {    // uniform per wave
        const int m0 = tileM0 + g * 8;
        const int b0 = 2 * m0 - 8;                 // absolute base sample

        // Gather B (all lanes, no divergence at the WMMAs below).
        v2f Bm[6];
        if (b0 >= lo && b0 + 23 <= hi) {
            // Interior: one base + immediate offsets -> ds_load offset: form.
            const int base = (b0 - lo + 2 * khi) * NCH + colm;
#pragma unroll
            for (int c = 0; c < 6; ++c) {
                Bm[c].x = smem[base + (4 * c)     * NCH];
                Bm[c].y = smem[base + (4 * c + 1) * NCH];
            }
        } else {
            // Edge: symmetric mirror + clamp into loaded span.
#pragma unroll
            for (int c = 0; c < 6; ++c) {
                const int k0 = 4 * c + 2 * khi;
                int i0 = mirror_idx(b0 + k0,     n) - lo;
                int i1 = mirror_idx(b0 + k0 + 1, n) - lo;
                i0 = (i0 < 0) ? 0 : (i0 > spanMax ? spanMax : i0);
                i1 = (i1 < 0) ? 0 : (i1 > spanMax ? spanMax : i1);
                Bm[c].x = smem[i0 * NCH + colm];
                Bm[c].y = smem[i1 * NCH + colm];
            }
        }

        v8f acc = {0.f, 0.f, 0.f, 0.f, 0.f, 0.f, 0.f, 0.f};
#pragma unroll
        for (int c = 0; c < 6; ++c) {
            acc = __builtin_amdgcn_wmma_f32_16x16x4_f32(
                false, Amat[c], false, Bm[c], (short)0, acc, false, false);
        }

        // D layout: lane = column (channel), VGPR v -> row r = v + 8*khi = 2p+f.
        if (m0 + 7 < nout) {
            // Full group: transpose through wave-private LDS, then two fully
            // coalesced 256B global_store_b128 (cA block + cD block).
            if (col < NCH) {
#pragma unroll
                for (int v = 0; v < 8; ++v) {
                    const int f = v & 1;
                    const int ml = (v >> 1) + 4 * khi;         // position in group
                    stage[wav][f * 64 + ml * NCH + col] = acc[v];
                }
            }
            // Same-wave LDS ops are in-order: read back other lanes' writes.
            const v4f vout = ((const v4f*)&stage[wav][0])[lane];
            const int rem = (lane & 15) * 4;                   // float offset in block
            if (lane < 16) {
                *(v4f*)(outA + (size_t)m0 * NCH + rem) = vout;             // cA: L2-hot
            } else {
                __builtin_nontemporal_store(vout, (v4f*)(outD + (size_t)m0 * NCH + rem));
            }
        } else {
            // Edge group: per-element bounds-checked scalar stores (rare).
            if (col < NCH) {
#pragma unroll
                for (int v = 0; v < 8; ++v) {
                    const int m = m0 + ((v + 8 * khi) >> 1);
                    if (m < nout) {
                        float* dst = (v & 1) ? outD : outA;
                        dst[(size_t)m * NCH + col] = acc[v];
                    }
                }
            }
        }
    }
#else
    // VALU fallback: (position, channel-half) items, packed f32 FMAs.
    const v4f* s4 = (const v4f*)smem;
    const int items = (mEnd - tileM0) * 2;
    for (int j = threadIdx.x; j < items; j += blockDim.x) {
        const int m    = tileM0 + (j >> 1);
        const int half = j & 1;
        v4f accA = {0.f, 0.f, 0.f, 0.f};
        v4f accD = {0.f, 0.f, 0.f, 0.f};
#pragma unroll
        for (int k = 0; k < 10; ++k) {
            int t = mirror_idx(2 * m + 1 - k, n);
            v4f v = s4[(t - lo) * 2 + half];
            accA += c_LO[k] * v;
            accD += c_HI[k] * v;
        }
        __builtin_nontemporal_store(accA, (v4f*)(outA + (size_t)m * NCH) + half);
        __builtin_nontemporal_store(accD, (v4f*)(outD + (size_t)m * NCH) + half);
    }
#endif
}

extern "C" void kernel_launch(void* const* d_in, const int* in_sizes, int n_in,
                              void* d_out, int out_size, void* d_ws, size_t ws_size,
                              hipStream_t stream) {
    const float* x   = (const float*)d_in[0];
    float*       out = (float*)d_out;

    const long N = (long)in_sizes[0] / NCH;            // 4194304

    // 1) copy orig to the front of the output
    {
        const long n4 = N * NCH / 4;
        copy_orig_kernel<<<8192, 256, 0, stream>>>((const v4f*)x, (v4f*)out, n4);
    }

    // Per-level lengths: n_out = (n+7)/2 + 1
    const int n0 = (int)N;
    const int n1 = (n0 + 7) / 2 + 1;                   // 2097156
    const int n2 = (n1 + 7) / 2 + 1;                   // 1048582
    const int n3 = (n2 + 7) / 2 + 1;                   // 524295
    const int n4 = (n3 + 7) / 2 + 1;                   // 262152

    // Output section offsets (positions): [cA4, cD4, cD3, cD2, cD1]
    const long offcA4 = 0;
    const long offcD4 = offcA4 + n4;
    const long offcD3 = offcD4 + n4;
    const long offcD2 = offcD3 + n3;
    const long offcD1 = offcD2 + n2;

    float* wave = out + (size_t)N * NCH;               // interleaved [pos][8]

    // Scratch for cA intermediates (cA3 reuses cA1's slot; cA1 dead by then).
    float* cA1 = (float*)d_ws;
    float* cA2 = cA1 + (size_t)n1 * NCH;
    float* cA3 = (float*)d_ws;

    const int blk = 256;
    dwt_level_kernel<<<(n1 + TILE - 1) / TILE, blk, 0, stream>>>(
        x,   n0, cA1,                 wave + offcD1 * NCH, n1);
    dwt_level_kernel<<<(n2 + TILE - 1) / TILE, blk, 0, stream>>>(
        cA1, n1, cA2,                 wave + offcD2 * NCH, n2);
    dwt_level_kernel<<<(n3 + TILE - 1) / TILE, blk, 0, stream>>>(
        cA2, n2, cA3,                 wave + offcD3 * NCH, n3);
    dwt_level_kernel<<<(n4 + TILE - 1) / TILE, blk, 0, stream>>>(
        cA3, n3, wave + offcA4 * NCH, wave + offcD4 * NCH, n4);
}